// RelativePositionEncoding_18373870092380
// MI455X (gfx1250) — compile-verified
//
#include <hip/hip_runtime.h>
#include <stdint.h>

// ---------------------------------------------------------------------------
// RelativePositionEncoding for MI455X (gfx1250).
//
// Store-bandwidth-bound op: 512 MB of f32 output, ~22.5 us floor at 23.3 TB/s.
// Strategy: per-block async DMA (global_load_async_to_lds_b128, ASYNCcnt) of
// the 69.6 KB weight table + 20 KB index arrays into LDS, then a gather-and-
// sum hot loop reading only LDS and emitting coalesced 512 B non-temporal
// b128 stores. WMMA is deliberately NOT used: the dense one-hot matmul
// equivalent (36.5 GFLOP of v_wmma_f32_16x16x4_f32) costs far more time than
// the store wall and cannot help a memory-bound kernel.
// ---------------------------------------------------------------------------

typedef float v4f __attribute__((ext_vector_type(4)));

#define NTOK    1024
#define CZ      128
#define RMAXI   32
#define SMAXI   2
#define RBINS   65            // 2*RMAX+1
#define WROWS   136           // 65 + 65 + 1 + 5
#define WELEM   (WROWS * CZ)  // 17408 f32 = 69632 B
#define THREADS 256
#define WAVES   8

// Low 32 bits of a flat pointer to __shared__ = LDS byte offset (aperture
// check uses addr[63:32]).
__device__ __forceinline__ uint32_t lds_addr32(const void* p) {
  return (uint32_t)(uintptr_t)p;
}

__device__ __forceinline__ void async_copy_b128(uint32_t lds_byte_addr,
                                                uint64_t global_addr) {
  // GV-mode async global->LDS, 16 B per lane. Tracked by ASYNCcnt.
  asm volatile("global_load_async_to_lds_b128 %0, %1, off"
               :: "v"(lds_byte_addr), "v"(global_addr)
               : "memory");
}

__device__ __forceinline__ int clampi(int v, int lo, int hi) {
  v = v < lo ? lo : v;
  return v > hi ? hi : v;
}

__global__ __launch_bounds__(THREADS)
void relpos_enc_kernel(const int* __restrict__ asym,
                       const int* __restrict__ resi,
                       const int* __restrict__ enti,
                       const int* __restrict__ toki,
                       const int* __restrict__ symi,
                       const float* __restrict__ W,
                       float* __restrict__ out) {
  __shared__ float sW[WELEM];        // 69632 B: full weight table
  __shared__ int   sIdx[5 * NTOK];   // 20480 B: asym|resi|enti|toki|symi

  const int tid  = threadIdx.x;        // 0..255
  const int lane = tid & 31;           // channel group: lane*4 .. lane*4+3
  const int wv   = tid >> 5;           // wave id 0..7
  const int i    = blockIdx.x;         // row index

  // ---- Async DMA stage: W table, 4352 x b128 -> 17 per thread ----
  {
    uint64_t g = (uint64_t)(uintptr_t)W + (uint64_t)tid * 16u;
    uint32_t l = lds_addr32(sW) + (uint32_t)tid * 16u;
#pragma unroll
    for (int k = 0; k < WELEM / 4 / THREADS; ++k) {  // 17 iterations
      async_copy_b128(l, g);
      g += (uint64_t)THREADS * 16u;
      l += THREADS * 16u;
    }
  }
  // ---- Async DMA stage: five 1024-int index arrays (1 b128 per thread) ----
  {
    const int* srcs[5] = {asym, resi, enti, toki, symi};
#pragma unroll
    for (int a = 0; a < 5; ++a) {
      uint64_t g = (uint64_t)(uintptr_t)srcs[a] + (uint64_t)tid * 16u;
      uint32_t l = lds_addr32(&sIdx[a * NTOK]) + (uint32_t)tid * 16u;
      async_copy_b128(l, g);
    }
  }
  // Drain this wave's ASYNCcnt, then workgroup barrier -> LDS fully populated.
  asm volatile("s_wait_asynccnt 0x0" ::: "memory");
  __syncthreads();

  // i-side scalars (block-uniform -> scalar loads, hit L2 across blocks).
  const int ai = asym[i];
  const int ri = resi[i];
  const int ei = enti[i];
  const int ti = toki[i];
  const int si = symi[i];

  const v4f* s4 = reinterpret_cast<const v4f*>(sW);
  // j-invariant same_entity row (W[2*RBINS] = row 130): hoisted.
  const v4f went = s4[130 * (CZ / 4) + lane];

  v4f* out4 = reinterpret_cast<v4f*>(out) + (size_t)i * NTOK * (CZ / 4);

  for (int j = wv; j < NTOK; j += WAVES) {
    const int aj = sIdx[0 * NTOK + j];
    const int rj = sIdx[1 * NTOK + j];
    const int ej = sIdx[2 * NTOK + j];
    const int tj = sIdx[3 * NTOK + j];
    const int sj = sIdx[4 * NTOK + j];

    const bool same_chain   = (aj == ai);
    const bool same_residue = (rj == ri);

    // rel_pos bin (row 0..64)
    int r0 = same_chain ? (2 * RMAXI)
                        : clampi(rj - ri + RMAXI, 0, 2 * RMAXI);
    // rel_token bin (row 65..129)
    int r1 = (same_residue || same_chain)
                 ? (2 * RMAXI)
                 : clampi(tj - ti + RMAXI, 0, 2 * RMAXI);
    r1 += RBINS;
    // rel_chain bin (row 131..135)
    int r3 = same_chain ? (2 * SMAXI)
                        : clampi(sj - si + SMAXI, 0, 2 * SMAXI);
    r3 += 2 * RBINS + 1;

    const float ef = (ej == ei) ? 1.0f : 0.0f;

    // Three gathered b128 LDS reads (lane-consecutive: conflict-free).
    const v4f a = s4[r0 * (CZ / 4) + lane];
    const v4f b = s4[r1 * (CZ / 4) + lane];
    const v4f d = s4[r3 * (CZ / 4) + lane];

    v4f o = a + b + d;
    o += ef * went;  // fma with the masked entity row

    // One coalesced 512 B line per wave; output (512 MB) >> L2 (192 MB),
    // so stream it with a non-temporal store.
    __builtin_nontemporal_store(o, &out4[(size_t)j * (CZ / 4) + lane]);
  }
}

extern "C" void kernel_launch(void* const* d_in, const int* in_sizes, int n_in,
                              void* d_out, int out_size, void* d_ws, size_t ws_size,
                              hipStream_t stream) {
  (void)in_sizes; (void)n_in; (void)out_size; (void)d_ws; (void)ws_size;
  const int*   asym = (const int*)d_in[0];
  const int*   resi = (const int*)d_in[1];
  const int*   enti = (const int*)d_in[2];
  const int*   toki = (const int*)d_in[3];
  const int*   symi = (const int*)d_in[4];
  const float* W    = (const float*)d_in[5];
  float*       out  = (float*)d_out;

  dim3 grid(NTOK);      // one block per i
  dim3 block(THREADS);  // 8 wave32s
  relpos_enc_kernel<<<grid, block, 0, stream>>>(asym, resi, enti, toki, symi, W, out);
}